// GAT_82815559402093
// MI455X (gfx1250) — compile-verified
//
#include <hip/hip_runtime.h>

#define NB 4
#define NN 10000
#define ND 64
#define NC 2
#define NE 160000
#define GAT_EPS 1e-10f

typedef __attribute__((ext_vector_type(2))) float v2f;
typedef __attribute__((ext_vector_type(8))) float v8f;

// ---------------------------------------------------------------- zero init
__global__ __launch_bounds__(256) void gat_zero(float* __restrict__ p, long long n) {
    long long i = (long long)blockIdx.x * blockDim.x + threadIdx.x;
    if (i < n) p[i] = 0.0f;
}

// ------------------------------------------------- GEMV via WMMA f32 16x16x4
// s[b][n][col], col = 2*c + half :
//   col0 = x·wa[c0,0:64], col1 = x·wa[c0,64:128], col2/3 = same for c1.
// One wave handles a 16-node tile; K=64 -> 16 chained V_WMMA_F32_16X16X4_F32.
// B-operand loads are unconditional (clamped address) + select-to-zero so the
// K-loop is branch-free and EXEC stays all-ones throughout.
__global__ __launch_bounds__(256) void gat_score_wmma(const float* __restrict__ x,
                                                      const float* __restrict__ wa,
                                                      float* __restrict__ s) {
    const int lane  = threadIdx.x & 31;
    const int wave  = threadIdx.x >> 5;
    const int tile  = blockIdx.x * 8 + wave;
    const int tiles_per_b = NN / 16;                 // 625 (exact)
    if (tile >= NB * tiles_per_b) return;            // wave-uniform skip
    const int b     = tile / tiles_per_b;
    const int node0 = (tile % tiles_per_b) * 16;

    const int mrow  = lane & 15;   // A: node row M ; B/D: column N
    const int khalf = lane >> 4;   // lanes 0-15 hold K pair {0,1}, 16-31 hold {2,3}

    const float* xrow = x + ((size_t)b * NN + node0 + mrow) * ND;

    // B column for this lane: only columns 0..3 carry weights; clamp the
    // address so every lane loads (no exec divergence), zero via multiplier.
    const int   col  = mrow;
    const int   colc = col < 4 ? col : 3;
    const float wsel = col < 4 ? 1.0f : 0.0f;
    const float* wcol = wa + (size_t)(colc >> 1) * (2 * ND) + (size_t)(colc & 1) * ND;

    v8f acc = {};
    #pragma unroll
    for (int it = 0; it < ND / 4; ++it) {
        const int k = it * 4 + khalf * 2;
        v2f a;  a.x  = xrow[k];         a.y  = xrow[k + 1];
        v2f bm; bm.x = wcol[k] * wsel;  bm.y = wcol[k + 1] * wsel;
        // (neg_a, A, neg_b, B, c_mod, C, reuse_a, reuse_b)
        acc = __builtin_amdgcn_wmma_f32_16x16x4_f32(false, a, false, bm,
                                                    (short)0, acc, false, false);
    }

    // D layout: VGPR r holds M = r + 8*khalf, column N = lane&15. Keep N<4.
    if (col < 4) {
        #pragma unroll
        for (int r8 = 0; r8 < 8; ++r8) {
            const int m = r8 + 8 * khalf;
            s[((size_t)b * NN + node0 + m) * 4 + col] = acc[r8];
        }
    }
}

// ------------------------------------------- pass 1: e = exp(lrelu), denom +=
__global__ __launch_bounds__(256) void gat_edge_pass1(const int*   __restrict__ eidx,
                                                      const float* __restrict__ s,
                                                      float*       __restrict__ ebuf,
                                                      float*       __restrict__ denom) {
    long long t = (long long)blockIdx.x * blockDim.x + threadIdx.x;
    const long long total = (long long)NB * NC * NE;
    if (t >= total) return;
    const long long bc = t / NE;
    const int c = (int)(bc % NC);
    const int b = (int)(bc / NC);
    const int* ep = eidx + (size_t)t * 2;
    const int i0 = ep[0];
    const int i1 = ep[1];
    const float score = s[((size_t)b * NN + i1) * 4 + 2 * c]
                      + s[((size_t)b * NN + i0) * 4 + 2 * c + 1];
    const float l  = score > 0.0f ? score : 0.2f * score;
    const float ev = __expf(l);
    ebuf[t] = ev;
    atomicAdd(&denom[(size_t)bc * NN + i0], ev);
}

// ---------------- pass 2: alpha = e/(denom[i1]+eps); r[i0,:] += alpha*x[i1,:]
// 16 lanes per edge, one float4 slice of the 64-dim row each.
__global__ __launch_bounds__(256) void gat_edge_pass2(const int*   __restrict__ eidx,
                                                      const float* __restrict__ x,
                                                      const float* __restrict__ ebuf,
                                                      const float* __restrict__ denom,
                                                      float*       __restrict__ r) {
    long long t = (long long)blockIdx.x * blockDim.x + threadIdx.x;
    const long long total = (long long)NB * NC * NE * 16;
    if (t >= total) return;
    const int sub = (int)(t & 15);        // which float4 of the 64-dim vector
    const long long eg = t >> 4;          // global edge id, (b,c,e) order
    const long long bc = eg / NE;
    const int b = (int)(bc / NC);
    const int* ep = eidx + (size_t)eg * 2;
    const int i0 = ep[0];
    const int i1 = ep[1];
    const float alpha = ebuf[eg] / (denom[(size_t)bc * NN + i1] + GAT_EPS);
    const float4 xv = *(const float4*)(x + ((size_t)b * NN + i1) * ND + sub * 4);
    float* rp = r + ((size_t)bc * NN + i0) * ND + sub * 4;
    atomicAdd(rp + 0, alpha * xv.x);
    atomicAdd(rp + 1, alpha * xv.y);
    atomicAdd(rp + 2, alpha * xv.z);
    atomicAdd(rp + 3, alpha * xv.w);
}

// ------------------------------------- finalize: out = sum_c sigmoid(r_c)
__global__ __launch_bounds__(256) void gat_finalize(const float* __restrict__ r,
                                                    float*       __restrict__ out) {
    long long t = (long long)blockIdx.x * blockDim.x + threadIdx.x;
    const long long total = (long long)NB * NN * ND;
    if (t >= total) return;
    const int d = (int)(t % ND);
    const long long bn = t / ND;
    const int n = (int)(bn % NN);
    const int b = (int)(bn / NN);
    const float r0 = r[(((size_t)(b * NC + 0) * NN + n) * ND) + d];
    const float r1 = r[(((size_t)(b * NC + 1) * NN + n) * ND) + d];
    out[t] = 1.0f / (1.0f + __expf(-r0)) + 1.0f / (1.0f + __expf(-r1));
}

extern "C" void kernel_launch(void* const* d_in, const int* in_sizes, int n_in,
                              void* d_out, int out_size, void* d_ws, size_t ws_size,
                              hipStream_t stream) {
    const float* x    = (const float*)d_in[0];   // (B, N, D)   f32
    const int*   eidx = (const int*)  d_in[1];   // (B, C, E, 2) i32
    const float* wa   = (const float*)d_in[2];   // (C, 2D, 1)  f32
    float* out = (float*)d_out;                  // (B, N, D)   f32

    // workspace layout (floats)
    float* ws    = (float*)d_ws;
    float* s     = ws;                                    // B*N*4      = 160,000
    float* denom = s     + (size_t)NB * NN * 4;           // B*C*N      =  80,000
    float* ebuf  = denom + (size_t)NB * NC * NN;          // B*C*E      = 1,280,000
    float* r     = ebuf  + (size_t)NB * NC * NE;          // B*C*N*D    = 5,120,000

    const int TPB = 256;

    // 1) zero the accumulators
    {
        long long nd = (long long)NB * NC * NN;
        gat_zero<<<(int)((nd + TPB - 1) / TPB), TPB, 0, stream>>>(denom, nd);
        long long nr = (long long)NB * NC * NN * ND;
        gat_zero<<<(int)((nr + TPB - 1) / TPB), TPB, 0, stream>>>(r, nr);
    }

    // 2) per-node scores via WMMA (fp32, exact): 2500 tiles, 8 waves/block
    {
        const int tiles = NB * (NN / 16);                 // 2500
        gat_score_wmma<<<(tiles + 7) / 8, TPB, 0, stream>>>(x, wa, s);
    }

    // 3) edge pass 1
    {
        long long n = (long long)NB * NC * NE;
        gat_edge_pass1<<<(int)((n + TPB - 1) / TPB), TPB, 0, stream>>>(eidx, s, ebuf, denom);
    }

    // 4) edge pass 2
    {
        long long n = (long long)NB * NC * NE * 16;
        gat_edge_pass2<<<(int)((n + TPB - 1) / TPB), TPB, 0, stream>>>(eidx, x, ebuf, denom, r);
    }

    // 5) finalize
    {
        long long n = (long long)NB * NN * ND;
        gat_finalize<<<(int)((n + TPB - 1) / TPB), TPB, 0, stream>>>(r, out);
    }
}